// GraphCutPostProcessor_90314572300379
// MI455X (gfx1250) — compile-verified
//
#include <hip/hip_runtime.h>
#include <hip/hip_bf16.h>
#include <float.h>

// ---------------------------------------------------------------------------
// GrabCut surrogate post-processor for MI455X (gfx1250, wave32).
// Memory-bound pipeline; deterministic two-stage reductions; WMMA f32
// 16x16x4 (exact f32 math) used as the final block-reduction stage.
// ---------------------------------------------------------------------------

#define BB 2
#define CC 256
#define HH 512
#define WW 512
#define HW (HH * WW)
#define CHW ((size_t)CC * HW)
#define MM_BLOCKS 256   // blocks per batch for min/max pass
#define NBLK 128        // blocks per batch for mean-reduction pass
#define GAMMA_W 50.0f

typedef __attribute__((ext_vector_type(2))) float v2f;
typedef __attribute__((ext_vector_type(8))) float v8f;

// trimap codes: 0=BGD 1=FGD 2=PR_BGD 3=PR_FGD
// bh=bw=int(512*0.1)=51 ; center box rows/cols [230,281)
__device__ __forceinline__ int trimap_of(int m, int y, int x) {
  int tri = (m == 1) ? 3 : 2;
  if (y < 51 || y >= HH - 51 || x < 51 || x >= WW - 51) tri = 0;
  if (y >= 230 && y < 281 && x >= 230 && x < 281) tri = 1;
  return tri;
}

// quantize like cv2.normalize(...,0,255,NORM_MINMAX) + floor + clip (all f32)
__device__ __forceinline__ float quant(float x, float mn, float denom) {
  float v = floorf((x - mn) / denom * 255.0f);
  return fminf(fmaxf(v, 0.0f), 255.0f);
}

// ---------------- pass 1: per-batch min/max over first 3 channels ----------
__global__ void k_minmax_partial(const float* __restrict__ f,
                                 float* __restrict__ part) {
  const int b = blockIdx.y, blk = blockIdx.x, tid = threadIdx.x;
  const float* base = f + (size_t)b * CHW;   // ch0,ch1,ch2 contiguous: 3*HW
  float mn = FLT_MAX, mx = -FLT_MAX;
  const int N = 3 * HW;
  for (int i = blk * 256 + tid; i < N; i += MM_BLOCKS * 256) {
    float v = base[i];
    mn = fminf(mn, v);
    mx = fmaxf(mx, v);
  }
  __shared__ float smn[256], smx[256];
  smn[tid] = mn; smx[tid] = mx;
  __syncthreads();
  for (int s = 128; s > 0; s >>= 1) {
    if (tid < s) {
      smn[tid] = fminf(smn[tid], smn[tid + s]);
      smx[tid] = fmaxf(smx[tid], smx[tid + s]);
    }
    __syncthreads();
  }
  if (tid == 0) {
    part[(b * MM_BLOCKS + blk) * 2 + 0] = smn[0];
    part[(b * MM_BLOCKS + blk) * 2 + 1] = smx[0];
  }
}

__global__ void k_minmax_final(const float* __restrict__ part,
                               float* __restrict__ minmax) {
  const int b = blockIdx.x, tid = threadIdx.x;  // 256 threads
  __shared__ float smn[256], smx[256];
  smn[tid] = part[(b * MM_BLOCKS + tid) * 2 + 0];
  smx[tid] = part[(b * MM_BLOCKS + tid) * 2 + 1];
  __syncthreads();
  for (int s = 128; s > 0; s >>= 1) {
    if (tid < s) {
      smn[tid] = fminf(smn[tid], smn[tid + s]);
      smx[tid] = fmaxf(smx[tid], smx[tid + s]);
    }
    __syncthreads();
  }
  if (tid == 0) {
    minmax[b * 2 + 0] = smn[0];
    minmax[b * 2 + 1] = smx[0];
  }
}

// ---------------- alpha seeding from trimap --------------------------------
__global__ void k_init_alpha(const int* __restrict__ mask,
                             float* __restrict__ alpha) {
  const int b = blockIdx.y;
  const int idx = blockIdx.x * 256 + threadIdx.x;
  const int y = idx >> 9, x = idx & (WW - 1);
  const int tri = trimap_of(mask[(size_t)b * HW + idx], y, x);
  alpha[(size_t)b * HW + idx] = (tri == 1 || tri == 3) ? 1.0f : 0.0f;
}

// ---------------- per-iteration masked color-sum reduction -----------------
// acc: [fgR,fgG,fgB,fgCnt, bgR,bgG,bgB,bgCnt]  — final 64->8 stage via WMMA.
__global__ void k_reduce(const float* __restrict__ f,
                         const float* __restrict__ alpha,
                         const float* __restrict__ minmax,
                         float* __restrict__ part) {
  const int b = blockIdx.y, blk = blockIdx.x, tid = threadIdx.x;
  const float mn = minmax[b * 2 + 0];
  const float denom = minmax[b * 2 + 1] - mn + 1e-12f;
  const float* r0 = f + (size_t)b * CHW;
  const float* g0 = r0 + HW;
  const float* b0 = r0 + 2 * HW;
  const float* al = alpha + (size_t)b * HW;

  float acc[8];
#pragma unroll
  for (int j = 0; j < 8; ++j) acc[j] = 0.0f;

  const int stride = NBLK * 256;
  for (int i = blk * 256 + tid; i < HW; i += stride) {
    if (i + stride < HW) {  // gfx1250 global_prefetch_b8 of next strided line
      __builtin_prefetch(&r0[i + stride], 0, 1);
      __builtin_prefetch(&al[i + stride], 0, 1);
    }
    const float a = al[i];
    const float rr = quant(r0[i], mn, denom);
    const float gg = quant(g0[i], mn, denom);
    const float bb = quant(b0[i], mn, denom);
    const float na = 1.0f - a;
    acc[0] += rr * a;  acc[1] += gg * a;  acc[2] += bb * a;  acc[3] += a;
    acc[4] += rr * na; acc[5] += gg * na; acc[6] += bb * na; acc[7] += na;
  }

  __shared__ float lds[256 * 8];
#pragma unroll
  for (int j = 0; j < 8; ++j) lds[tid * 8 + j] = acc[j];
  __syncthreads();
  // tree-reduce 256 -> 8 threads (deterministic order)
  for (int s = 128; s >= 8; s >>= 1) {
    if (tid < s) {
#pragma unroll
      for (int j = 0; j < 8; ++j) lds[tid * 8 + j] += lds[(tid + s) * 8 + j];
    }
    __syncthreads();
  }

  // Final stage: 8 partials x 8 accumulators = 64 f32 = one 16x4 A-matrix.
  // A[m][k] = partial p=(m&1)*4+k of accumulator j=m>>1 ; B = ones(4x16).
  // D[m][n] = rowsum_m, so acc_j = rowsum_{2j} + rowsum_{2j+1}.
  if (tid < 32) {
    const int m = tid & 15;
    const int j = m >> 1;
    const int kbase = (tid >> 4) * 2;          // lanes 16..31 carry K=2,3
    const int p0 = (m & 1) * 4 + kbase;
    v2f a;
    a[0] = lds[p0 * 8 + j];
    a[1] = lds[(p0 + 1) * 8 + j];
    v2f ones;
    ones[0] = 1.0f; ones[1] = 1.0f;
    v8f c = {};
    v8f d = __builtin_amdgcn_wmma_f32_16x16x4_f32(
        /*neg_a=*/false, a, /*neg_b=*/false, ones,
        /*c_mod=*/(short)0, c, /*reuse_a=*/false, /*reuse_b=*/false);
    // lanes 0..15: D vgpr r = rowsum_r   -> acc_i   = d[2i]+d[2i+1]
    // lanes 16..31: D vgpr r = rowsum_8+r -> acc_{4+i}= d[2i]+d[2i+1]
    float s0 = d[0] + d[1];
    float s1 = d[2] + d[3];
    float s2 = d[4] + d[5];
    float s3 = d[6] + d[7];
    float* po = part + ((size_t)b * NBLK + blk) * 8;
    if (tid == 0)  { po[0] = s0; po[1] = s1; po[2] = s2; po[3] = s3; }
    if (tid == 16) { po[4] = s0; po[5] = s1; po[6] = s2; po[7] = s3; }
  }
}

__global__ void k_means(const float* __restrict__ part,
                        float* __restrict__ means) {
  const int b = blockIdx.x, tid = threadIdx.x;  // 128 threads
  __shared__ float lds[NBLK * 8];
  const float* p = part + (size_t)b * NBLK * 8;
#pragma unroll
  for (int j = 0; j < 8; ++j) lds[tid * 8 + j] = p[tid * 8 + j];
  __syncthreads();
  for (int s = 64; s > 0; s >>= 1) {
    if (tid < s) {
#pragma unroll
      for (int j = 0; j < 8; ++j) lds[tid * 8 + j] += lds[(tid + s) * 8 + j];
    }
    __syncthreads();
  }
  if (tid == 0) {
    const float fgc = lds[3] + 1e-6f;
    const float bgc = lds[7] + 1e-6f;
    means[b * 8 + 0] = lds[0] / fgc;
    means[b * 8 + 1] = lds[1] / fgc;
    means[b * 8 + 2] = lds[2] / fgc;
    means[b * 8 + 4] = lds[4] / bgc;
    means[b * 8 + 5] = lds[5] / bgc;
    means[b * 8 + 6] = lds[6] / bgc;
  }
}

// ---------------- per-iteration ICM relabel (4-neighbor, edge clamp) -------
__global__ void k_relabel(const float* __restrict__ f,
                          const int* __restrict__ mask,
                          const float* __restrict__ asrc,
                          float* __restrict__ adst,
                          const float* __restrict__ minmax,
                          const float* __restrict__ means) {
  const int b = blockIdx.y;
  const int idx = blockIdx.x * 256 + threadIdx.x;
  const int y = idx >> 9, x = idx & (WW - 1);

  const float mn = minmax[b * 2 + 0];
  const float denom = minmax[b * 2 + 1] - mn + 1e-12f;
  const float fr = means[b * 8 + 0], fg = means[b * 8 + 1], fb = means[b * 8 + 2];
  const float br = means[b * 8 + 4], bg = means[b * 8 + 5], bbm = means[b * 8 + 6];

  const float* al = asrc + (size_t)b * HW;
  const int yu = max(y - 1, 0), yd = min(y + 1, HH - 1);
  const int xl = max(x - 1, 0), xr = min(x + 1, WW - 1);
  const float nb = 0.25f * (al[yu * WW + x] + al[yd * WW + x] +
                            al[y * WW + xl] + al[y * WW + xr]);

  const float* base = f + (size_t)b * CHW;
  const float rr = quant(base[idx], mn, denom);
  const float gg = quant(base[HW + idx], mn, denom);
  const float bb = quant(base[2 * HW + idx], mn, denom);

  const float dfr = rr - fr, dfgc = gg - fg, dfb = bb - fb;
  const float dbr = rr - br, dbgc = gg - bg, dbb = bb - bbm;
  const float d_fg = dfr * dfr + dfgc * dfgc + dfb * dfb;
  const float d_bg = dbr * dbr + dbgc * dbgc + dbb * dbb;

  const float score = d_bg - d_fg + GAMMA_W * (2.0f * nb - 1.0f);
  const float newa = (score > 0.0f) ? 1.0f : 0.0f;

  const int tri = trimap_of(mask[(size_t)b * HW + idx], y, x);
  const bool fixedpx = (tri == 0) || (tri == 1);
  adst[(size_t)b * HW + idx] = fixedpx ? ((tri == 1) ? 1.0f : 0.0f) : newa;
}

// ---------------------------------------------------------------------------
extern "C" void kernel_launch(void* const* d_in, const int* in_sizes, int n_in,
                              void* d_out, int out_size, void* d_ws, size_t ws_size,
                              hipStream_t stream) {
  (void)in_sizes; (void)n_in; (void)out_size; (void)ws_size;
  const float* feats = (const float*)d_in[0];
  const int* mask = (const int*)d_in[1];
  float* out = (float*)d_out;
  float* ws = (float*)d_ws;

  float* minmax  = ws + 0;                         // 2*B
  float* means   = ws + 4;                         // 8*B
  float* mmpart  = ws + 20;                        // B*MM_BLOCKS*2 = 1024
  float* sumpart = ws + 20 + BB * MM_BLOCKS * 2;   // B*NBLK*8 = 2048
  float* alphaA  = ws + 4096;                      // B*HW = 524288

  k_minmax_partial<<<dim3(MM_BLOCKS, BB), 256, 0, stream>>>(feats, mmpart);
  k_minmax_final<<<BB, 256, 0, stream>>>(mmpart, minmax);
  k_init_alpha<<<dim3(HW / 256, BB), 256, 0, stream>>>(mask, alphaA);

  float* cur = alphaA;
  float* nxt = out;
  for (int it = 0; it < 5; ++it) {
    k_reduce<<<dim3(NBLK, BB), 256, 0, stream>>>(feats, cur, minmax, sumpart);
    k_means<<<BB, NBLK, 0, stream>>>(sumpart, means);
    k_relabel<<<dim3(HW / 256, BB), 256, 0, stream>>>(feats, mask, cur, nxt,
                                                      minmax, means);
    float* t = cur; cur = nxt; nxt = t;
  }
  // 5 iterations: ws->out, out->ws, ws->out, out->ws, ws->out  => final in d_out
}